// NegLogPartialLikelihood_48249662603872
// MI455X (gfx1250) — compile-verified
//
#include <hip/hip_runtime.h>
#include <hip/hip_bf16.h>
#include <math.h>

// Cox negative log partial likelihood, N^2 risk-set formulation.
// sum_risk[j] = sum_i exp(lr_i) * [t_i >= t_j]  computed with
// V_WMMA_F32_16X16X4_F32 (full f32, bit-faithful masks), then
// loss = -sum_j (lr_j - log(sum_risk_j)) * ystatus_j / sum(ystatus).

#define JT_PER_BLOCK 4   // waves per block; each wave owns one 16-wide j tile
#define ISPLIT 8         // split of the i (risk-set) dimension across grid.y

typedef __attribute__((ext_vector_type(2))) float v2f;
typedef __attribute__((ext_vector_type(8))) float v8f;

// ---------------------------------------------------------------------------
// Kernel A: e[i] = exp(log_risk[i]); zero sum_risk and the two accumulators.
// ---------------------------------------------------------------------------
__global__ __launch_bounds__(256) void cox_prep(const float* __restrict__ log_risk,
                                                float* __restrict__ e,
                                                float* __restrict__ sum_risk,
                                                float* __restrict__ acc, int n) {
    int i = blockIdx.x * blockDim.x + threadIdx.x;
    if (i < 2) acc[i] = 0.0f;                 // {num_events, numerator}
    if (i < n) {
        e[i] = expf(log_risk[i]);
        sum_risk[i] = 0.0f;
    }
}

// ---------------------------------------------------------------------------
// Kernel B: masked column-sums via V_WMMA_F32_16X16X4_F32.
//   A[m][k] = e_{i0+k}            (replicated over M; lane half selects K pair)
//   B[k][n] = [t_{i0+k} >= t_{j0+n}]   (exact 0.0/1.0 mask)
//   C[0][n] accumulates sum_risk partial for j = j0 + n.
// 32-bit A 16x4 layout (ISA 7.12.2): V0 = K0 (lanes 0-15) / K2 (lanes 16-31),
// V1 = K1 / K3.  B 4x16 mirrors this with N = lane & 15.
// ---------------------------------------------------------------------------
__global__ __launch_bounds__(32 * JT_PER_BLOCK) void cox_sumrisk_wmma(
        const float* __restrict__ t, const float* __restrict__ e,
        float* __restrict__ sum_risk, int n) {
    const int lane = threadIdx.x & 31;
    const int wave = threadIdx.x >> 5;
    const int j0   = (blockIdx.x * JT_PER_BLOCK + wave) * 16;
    const int chunk = n / ISPLIT;
    const int ibeg  = blockIdx.y * chunk;
    const int iend  = ibeg + chunk;

    const float tj   = t[j0 + (lane & 15)];   // this lane's column's event time
    const int   koff = (lane >= 16) ? 2 : 0;  // lane half -> K pair {0,1} or {2,3}

    v8f acc = {0.f, 0.f, 0.f, 0.f, 0.f, 0.f, 0.f, 0.f};

    #pragma unroll 4
    for (int i = ibeg; i < iend; i += 4) {
        v2f ti = *(const v2f*)(t + i + koff);  // broadcast loads, L2-resident
        v2f a  = *(const v2f*)(e + i + koff);  // A: exp values (f32, exact)
        v2f b;
        b.x = (ti.x >= tj) ? 1.0f : 0.0f;      // B: risk-set mask (exact in f32)
        b.y = (ti.y >= tj) ? 1.0f : 0.0f;
        // D = A x B + C : 8 args (neg_a, A, neg_b, B, c_mod, C, reuse_a, reuse_b)
        acc = __builtin_amdgcn_wmma_f32_16x16x4_f32(
                  false, a, false, b, (short)0, acc, false, false);
    }

    // Row M=0 of C: lanes 0-15, VGPR 0 hold columns N=0..15.
    if (lane < 16) atomicAdd(&sum_risk[j0 + lane], acc[0]);
}

// ---------------------------------------------------------------------------
// Kernel C: per-j contribution + block tree reduction into the accumulators.
// ---------------------------------------------------------------------------
__global__ __launch_bounds__(256) void cox_finalize(const float* __restrict__ log_risk,
                                                    const float* __restrict__ ystatus,
                                                    const float* __restrict__ sum_risk,
                                                    float* __restrict__ acc, int n) {
    __shared__ float sEv[256];
    __shared__ float sNum[256];
    const int tid = threadIdx.x;
    const int i   = blockIdx.x * blockDim.x + tid;

    float ev = 0.0f, num = 0.0f;
    if (i < n) {
        ev  = ystatus[i];
        num = (log_risk[i] - logf(sum_risk[i])) * ev;
    }
    sEv[tid] = ev;
    sNum[tid] = num;
    __syncthreads();

    for (int s = 128; s > 0; s >>= 1) {
        if (tid < s) {
            sEv[tid]  += sEv[tid + s];
            sNum[tid] += sNum[tid + s];
        }
        __syncthreads();
    }
    if (tid == 0) {
        atomicAdd(&acc[0], sEv[0]);   // num_events
        atomicAdd(&acc[1], sNum[0]);  // numerator
    }
}

// ---------------------------------------------------------------------------
// Kernel D: loss = -numerator / num_events
// ---------------------------------------------------------------------------
__global__ void cox_write(const float* __restrict__ acc, float* __restrict__ out) {
    out[0] = -acc[1] / acc[0];
}

// ---------------------------------------------------------------------------
extern "C" void kernel_launch(void* const* d_in, const int* in_sizes, int n_in,
                              void* d_out, int out_size, void* d_ws, size_t ws_size,
                              hipStream_t stream) {
    const float* log_risk = (const float*)d_in[0];  // [N,1] f32
    const float* ytime    = (const float*)d_in[1];  // [N]   f32
    const float* ystatus  = (const float*)d_in[2];  // [N]   f32
    float* out = (float*)d_out;                     // scalar f32

    const int n = in_sizes[1];                      // N = 16384

    // ws layout (floats): [0..1] accumulators | [16..16+n) e | [16+n..16+2n) sum_risk
    float* acc      = (float*)d_ws;
    float* e        = acc + 16;
    float* sum_risk = e + n;

    const int tpb = 256;
    cox_prep<<<(n + tpb - 1) / tpb, tpb, 0, stream>>>(log_risk, e, sum_risk, acc, n);

    dim3 gridB(n / 16 / JT_PER_BLOCK, ISPLIT);      // (256, 8) blocks of 4 waves
    cox_sumrisk_wmma<<<gridB, 32 * JT_PER_BLOCK, 0, stream>>>(ytime, e, sum_risk, n);

    cox_finalize<<<(n + tpb - 1) / tpb, tpb, 0, stream>>>(log_risk, ystatus, sum_risk, acc, n);
    cox_write<<<1, 1, 0, stream>>>(acc, out);
}